// GetCovMatFromChol_Sequence_54941221650672
// MI455X (gfx1250) — compile-verified
//
#include <hip/hip_runtime.h>

// Streaming, bandwidth-bound kernel for Q = L @ L^T with 3x3 lower-triangular L
// packed as 6 floats per element. MI455X (gfx1250), wave32.
//
// Per element: in = [a,b,c,d,e,f] ->
//   L = [[a,0,0],[b,c,0],[d,e,f]]
//   Q00=a*a  Q01=a*b        Q02=a*d
//   Q10=Q01  Q11=b*b+c*c    Q12=b*d+c*e
//   Q20=Q02  Q21=Q12        Q22=d*d+e*e+f*f
//
// Roofline: 60 B moved / 14 flops per element -> pure HBM-bound (~252 MB total,
// ~10.8 us floor at 23.3 TB/s). No WMMA: independent 3x3 products don't map to
// 16x16xK tiles. Optimization = NT b128 traffic + LDS-staged coalesced stores.

typedef float f2 __attribute__((ext_vector_type(2)));
typedef float f4 __attribute__((ext_vector_type(4)));

#define TPB 256        // 8 waves (wave32)
#define EPT 4          // elements per thread
#define EPB (TPB*EPT)  // 1024 elements per block

__device__ __forceinline__ void compute_and_stage(float* __restrict__ sQ,
                                                  int el, f2 p0, f2 p1, f2 p2) {
  const float a = p0.x, b = p0.y, c = p1.x;
  const float d = p1.y, e = p2.x, f = p2.y;

  const float q00 = a * a;
  const float q01 = a * b;
  const float q02 = a * d;
  const float q11 = __builtin_fmaf(b, b, c * c);
  const float q12 = __builtin_fmaf(b, d, c * e);
  const float q22 = __builtin_fmaf(d, d, __builtin_fmaf(e, e, f * f));

  // 9-dword stride across lanes: gcd(9,64)=1 -> bank-conflict-free (wave32).
  float* s = sQ + el * 9;
  s[0] = q00; s[1] = q01; s[2] = q02;
  s[3] = q01; s[4] = q11; s[5] = q12;
  s[6] = q02; s[7] = q12; s[8] = q22;
}

__global__ __launch_bounds__(TPB) void cov_from_chol_kernel(
    const float* __restrict__ in, float* __restrict__ out, int n_elems) {
  // Stage the 36-byte-stride output through LDS so global stores are
  // contiguous 16B-aligned b128. 1024 elems * 9 floats = 36 KB (<320 KB/WGP,
  // 8 blocks resident -> 16 waves/SIMD of latency hiding).
  __shared__ float sQ[EPB * 9];

  const int    t    = threadIdx.x;
  const size_t base = (size_t)blockIdx.x * EPB;
  const bool   full = (base + EPB) <= (size_t)n_elems;

  if (full) {
    // Branch-free fast path: issue all 8 vmem loads as one clause, then
    // compute — no per-element exec-mask dance, no per-element loadcnt wait.
    f2 r[EPT][3];
#pragma unroll
    for (int k = 0; k < EPT; ++k) {
      // 24-byte element stride -> always 8-byte aligned b64/b128 NT loads.
      const f2* p = (const f2*)(in + (base + (size_t)k * TPB + t) * 6);
      r[k][0] = __builtin_nontemporal_load(p + 0);
      r[k][1] = __builtin_nontemporal_load(p + 1);
      r[k][2] = __builtin_nontemporal_load(p + 2);
    }
#pragma unroll
    for (int k = 0; k < EPT; ++k) {
      compute_and_stage(sQ, k * TPB + t, r[k][0], r[k][1], r[k][2]);
    }
  } else {
    // Guarded tail path (unused for the 4096x1024 harness shape).
#pragma unroll
    for (int k = 0; k < EPT; ++k) {
      const int    el = k * TPB + t;
      const size_t e  = base + el;
      if (e < (size_t)n_elems) {
        const f2* p = (const f2*)(in + e * 6);
        f2 p0 = __builtin_nontemporal_load(p + 0);
        f2 p1 = __builtin_nontemporal_load(p + 1);
        f2 p2 = __builtin_nontemporal_load(p + 2);
        compute_and_stage(sQ, el, p0, p1, p2);
      }
    }
  }

  __syncthreads();

  // Coalesced write-back: 2304 float4 chunks per block, 9 per thread,
  // 16B-aligned (block output base = blockIdx * 36864 bytes), NT stores.
  const size_t ob4  = (size_t)blockIdx.x * (EPB * 9 / 4);
  f4*       out4 = (f4*)out;
  const f4* s4   = (const f4*)sQ;

  if (full) {
#pragma unroll
    for (int i = 0; i < (EPB * 9) / (4 * TPB); ++i) {
      const size_t idx = (size_t)i * TPB + t;
      __builtin_nontemporal_store(s4[idx], out4 + ob4 + idx);
    }
  } else {
    const size_t tot4 = ((size_t)n_elems * 9) / 4;
#pragma unroll
    for (int i = 0; i < (EPB * 9) / (4 * TPB); ++i) {
      const size_t idx = (size_t)i * TPB + t;
      const size_t g   = ob4 + idx;
      if (g < tot4) {
        __builtin_nontemporal_store(s4[idx], out4 + g);
      }
    }
  }
}

extern "C" void kernel_launch(void* const* d_in, const int* in_sizes, int n_in,
                              void* d_out, int out_size, void* d_ws, size_t ws_size,
                              hipStream_t stream) {
  const float* in  = (const float*)d_in[0];
  float*       out = (float*)d_out;
  const int n_elems = in_sizes[0] / 6;           // 4096*1024 = 4,194,304
  const int grid    = (n_elems + EPB - 1) / EPB; // 4096 blocks, all full tiles
  hipLaunchKernelGGL(cov_from_chol_kernel, dim3(grid), dim3(TPB), 0, stream,
                     in, out, n_elems);
}